// _RefVisionPooler_25159918420628
// MI455X (gfx1250) — compile-verified
//
#include <hip/hip_runtime.h>
#include <cstdint>

// RefVisionPooler for MI455X (gfx1250).
// Shapes fixed by the reference harness: B=8, S=4096, D=1152, OUT=1024,
// kernel_size = int(sqrt(S/OUT)) = 2.  Memory-bound: ~151MB read + ~38MB
// write -> ~8.1us floor at 23.3 TB/s.  Segment mapping is computed from the
// actual pixel_position_ids / padding inputs via a counting sort, so the
// bandwidth-critical pass is a pure streaming gather using CDNA5 async
// global->LDS DMA (GLOBAL_LOAD_ASYNC_TO_LDS_B128, ASYNCcnt), burst-issued
// 4 rows at a time with a single s_wait_asynccnt + barrier per chunk.

namespace {

constexpr int B    = 8;
constexpr int S    = 4096;
constexpr int D    = 1152;
constexpr int OUT  = 1024;
constexpr int K    = 2;              // int(sqrt(S/OUT))
constexpr int NBUF = 4;              // LDS row buffers (typ. cnt == K*K == 4)
constexpr int THREADS_POOL = D / 4;  // 288 lanes = 9 wave32s, one float4 each

// Generic pointers into LDS carry the LDS byte offset in their low 32 bits
// (aperture bits live in [63:32] per the CDNA5 flat-address aperture check).
__device__ __forceinline__ uint32_t lds_offset(const void* p) {
  return (uint32_t)(uintptr_t)p;
}

__global__ void zero_counts(int* counts, int n) {
  int i = blockIdx.x * blockDim.x + threadIdx.x;
  if (i < n) counts[i] = 0;
}

// Per-batch max_x = max(clamp(pos_x, 0)) + 1
__global__ void maxx_kernel(const int* __restrict__ pos, int* __restrict__ maxx) {
  __shared__ int red[256];
  const int b = blockIdx.x;
  int m = 0;
  for (int s = threadIdx.x; s < S; s += blockDim.x) {
    int x = pos[((size_t)b * S + s) * 2 + 0];
    m = max(m, max(x, 0));
  }
  red[threadIdx.x] = m;
  __syncthreads();
  for (int off = 128; off > 0; off >>= 1) {
    if (threadIdx.x < off)
      red[threadIdx.x] = max(red[threadIdx.x], red[threadIdx.x + off]);
    __syncthreads();
  }
  if (threadIdx.x == 0) maxx[b] = red[0] + 1;
}

__device__ __forceinline__ int seg_index(const int* __restrict__ pos,
                                         const int* __restrict__ maxx,
                                         int b, int s) {
  size_t base = ((size_t)b * S + s) * 2;
  int x = max(pos[base + 0], 0);
  int y = max(pos[base + 1], 0);
  return (x / K) + (maxx[b] / K) * (y / K);
}

__global__ void count_kernel(const int* __restrict__ pos, const int* __restrict__ maxx,
                             int* __restrict__ counts) {
  int i = blockIdx.x * blockDim.x + threadIdx.x;
  if (i >= B * S) return;
  int b = i / S, s = i % S;
  int idx = seg_index(pos, maxx, b, s);
  if (idx >= 0 && idx < OUT) atomicAdd(&counts[b * OUT + idx], 1);
}

// One block (OUT threads) per batch: exclusive scan of counts -> offsets/cursors.
__global__ void scan_kernel(const int* __restrict__ counts, int* __restrict__ offsets,
                            int* __restrict__ cursors) {
  __shared__ int sd[OUT];
  const int b = blockIdx.x, t = threadIdx.x;
  int v = counts[b * OUT + t];
  sd[t] = v;
  __syncthreads();
  for (int off = 1; off < OUT; off <<= 1) {
    int x = (t >= off) ? sd[t - off] : 0;
    __syncthreads();
    sd[t] += x;
    __syncthreads();
  }
  int excl = sd[t] - v;
  offsets[b * OUT + t] = excl;
  cursors[b * OUT + t] = excl;
}

__global__ void fill_kernel(const int* __restrict__ pos, const int* __restrict__ maxx,
                            int* __restrict__ cursors, int* __restrict__ perm) {
  int i = blockIdx.x * blockDim.x + threadIdx.x;
  if (i >= B * S) return;
  int b = i / S, s = i % S;
  int idx = seg_index(pos, maxx, b, s);
  if (idx >= 0 && idx < OUT) {
    int slot = atomicAdd(&cursors[b * OUT + idx], 1);
    perm[(size_t)b * S + slot] = s;
  }
}

// Bandwidth pass: one block per output row (B*OUT = 8192 blocks).
// Each lane owns one 16-byte feature chunk.  Rows are DMA'd into LDS with
// async b128 copies, NBUF rows burst-issued per chunk -> one s_wait_asynccnt
// and two barriers per chunk (vs. per-row in a depth-2 pipeline).
__global__ __launch_bounds__(THREADS_POOL) void pool_gather(
    const float* __restrict__ hs, const unsigned char* __restrict__ pad,
    const int* __restrict__ counts, const int* __restrict__ offsets,
    const int* __restrict__ perm, float* __restrict__ out,
    float* __restrict__ mask) {
  __shared__ __align__(16) float buf[NBUF][D];
  const int p = blockIdx.x;       // b*OUT + o
  const int b = p / OUT;
  const int t = threadIdx.x;
  const int cnt = counts[p];
  const int start = offsets[p];
  const float scale = sqrtf((float)D) / (float)(K * K); // sqrt(D)/k^2

  float ax = 0.f, ay = 0.f, az = 0.f, aw = 0.f;

  for (int base = 0; base < cnt; base += NBUF) {
    const int m = min(NBUF, cnt - base);

    // Hoist row ids + padding weights for this chunk, then burst-issue the
    // async copies (ASYNCcnt tracks up to 63 in flight; loads retire in order).
    int   srow[NBUF];
    float w[NBUF];
    for (int j = 0; j < m; ++j) {
      srow[j] = perm[(size_t)b * S + start + base + j];
      w[j]    = pad[(size_t)b * S + srow[j]] ? 0.0f : 1.0f; // hs zeroed at pad
    }
    for (int j = 0; j < m; ++j) {
      const float* src =
          hs + ((size_t)b * S + srow[j]) * (size_t)D + (size_t)t * 4;
      uint32_t lds = lds_offset(&buf[j][t * 4]);
      // CDNA5 async copy: 16B per lane, global -> LDS, tracked by ASYNCcnt.
      asm volatile("global_load_async_to_lds_b128 %0, %1, off"
                   :
                   : "v"(lds), "v"((uint64_t)(uintptr_t)src)
                   : "memory");
    }

    asm volatile("s_wait_asynccnt 0x0" ::: "memory"); // chunk's DMAs complete
    __syncthreads();                                  // visible to all 9 waves

    for (int j = 0; j < m; ++j) {
      const float4 v = *reinterpret_cast<const float4*>(&buf[j][t * 4]);
      ax += w[j] * v.x; ay += w[j] * v.y;
      az += w[j] * v.z; aw += w[j] * v.w;
    }

    if (base + NBUF < cnt) __syncthreads(); // WAR before buffers are reused
  }

  float4 r;
  r.x = ax * scale; r.y = ay * scale; r.z = az * scale; r.w = aw * scale;
  reinterpret_cast<float4*>(out + (size_t)p * D)[t] = r; // global_store_b128
  if (t == 0) mask[p] = (cnt > 0) ? 1.0f : 0.0f;
}

} // namespace

extern "C" void kernel_launch(void* const* d_in, const int* in_sizes, int n_in,
                              void* d_out, int out_size, void* d_ws, size_t ws_size,
                              hipStream_t stream) {
  (void)in_sizes; (void)n_in; (void)out_size; (void)ws_size;
  const float*         hs  = (const float*)d_in[0];
  const int*           pos = (const int*)d_in[1];
  const unsigned char* pad = (const unsigned char*)d_in[2];
  // d_in[3] = output_length scalar; shapes are fixed by the harness.

  // Workspace layout (ints): maxx[B] | counts[B*OUT] | offsets[B*OUT] |
  // cursors[B*OUT] | perm[B*S]  (~230 KB total). Re-initialized every call.
  int* ws     = (int*)d_ws;
  int* maxx   = ws;
  int* counts = maxx + B;
  int* offs   = counts + B * OUT;
  int* curs   = offs + B * OUT;
  int* perm   = curs + B * OUT;

  float* out  = (float*)d_out;
  float* mask = out + (size_t)B * OUT * D;

  zero_counts<<<(B * OUT + 255) / 256, 256, 0, stream>>>(counts, B * OUT);
  maxx_kernel<<<B, 256, 0, stream>>>(pos, maxx);
  count_kernel<<<(B * S + 255) / 256, 256, 0, stream>>>(pos, maxx, counts);
  scan_kernel<<<B, OUT, 0, stream>>>(counts, offs, curs);
  fill_kernel<<<(B * S + 255) / 256, 256, 0, stream>>>(pos, maxx, curs, perm);
  pool_gather<<<B * OUT, THREADS_POOL, 0, stream>>>(hs, pad, counts, offs, perm,
                                                    out, mask);
}